// snn_mnist_31267361915318
// MI455X (gfx1250) — compile-verified
//
#include <hip/hip_runtime.h>
#include <math.h>

// SNN (snntorch Synaptic LIF + STDP), T=350 serialized steps.
// fp32 WMMA (V_WMMA_F32_16X16X4_F32) for all three GEMMs: exact arithmetic,
// matrix pipe utilization; problem is sync-latency bound, not FLOP bound.

typedef float v2f __attribute__((ext_vector_type(2)));
typedef float v8f __attribute__((ext_vector_type(8)));

#define T_STEPS 350
#define BATCH   128
#define IN_DIM  784
#define N_NEUR  400

#define ALPHA      0.9f
#define BETA       0.8f
#define BETA_PLUS  0.9f
#define BETA_MINUS 0.9f
#define A_PLUS     0.008f
#define A_MINUS    0.0008f

__device__ __forceinline__ v8f wmma_f32_k4(v2f a, v2f b, v8f c) {
  // 8 args: (neg_a, A, neg_b, B, c_mod, C, reuse_a, reuse_b)
  return __builtin_amdgcn_wmma_f32_16x16x4_f32(false, a, false, b,
                                               (short)0, c, false, false);
}

// ---------------------------------------------------------------------------
// init: W master copy -> d_out, zero mem/syn/counts in d_out, zero traces+spk in ws
// ---------------------------------------------------------------------------
__global__ __launch_bounds__(256) void init_kernel(const float* __restrict__ W0,
                                                   float* __restrict__ out,
                                                   float* __restrict__ ws) {
  int gi = blockIdx.x * 256 + threadIdx.x;
  if (gi < N_NEUR * IN_DIM) {
    out[gi] = W0[gi];                       // W_final region
  } else if (gi < N_NEUR * IN_DIM + 2 * BATCH * N_NEUR + N_NEUR) {
    out[gi] = 0.0f;                         // mem, syn, spike counts
  }
  if (gi < BATCH * IN_DIM + 2 * BATCH * N_NEUR) {
    ws[gi] = 0.0f;                          // pre_tr, post_tr, spk
  }
}

// ---------------------------------------------------------------------------
// forward: blocks [0,50): I = x·Wᵀ (200 16x16 tiles, one per wave) + LIF update
//          blocks [50,442): pre_tr = beta_plus*pre_tr + x   (392*256 == 128*784)
// ---------------------------------------------------------------------------
__global__ __launch_bounds__(256) void fwd_kernel(
    const float* __restrict__ x, const float* __restrict__ W,
    float* __restrict__ syn, float* __restrict__ mem,
    float* __restrict__ post_tr, float* __restrict__ spk,
    float* __restrict__ pre_tr, float* __restrict__ counts) {
  if (blockIdx.x >= 50) {
    int idx = (blockIdx.x - 50) * 256 + threadIdx.x;   // exactly covers 100352
    pre_tr[idx] = BETA_PLUS * pre_tr[idx] + x[idx];
    return;
  }
  const int wave = threadIdx.x >> 5;
  const int lane = threadIdx.x & 31;
  const int tile = blockIdx.x * 4 + wave;   // 0..203 (8 waves/block, stride 4)
  const int m0 = (tile & 7) << 4;           // batch tile base (128/16 = 8)
  const int n0 = (tile >> 3) << 4;          // neuron tile base (400/16 = 25)
  const int r    = lane & 15;
  const int half = lane >> 4;

  v8f acc = {0.f, 0.f, 0.f, 0.f, 0.f, 0.f, 0.f, 0.f};
  const float* arow = x + (m0 + r) * IN_DIM;   // A[m][k] = x[m*IN + k]
  const float* brow = W + (n0 + r) * IN_DIM;   // B[k][n] = W[n*IN + k]
#pragma unroll 4
  for (int k = 0; k < IN_DIM; k += 4) {
    const int kk = k + 2 * half;
    v2f a = { arow[kk], arow[kk + 1] };
    v2f b = { brow[kk], brow[kk + 1] };
    acc = wmma_f32_k4(a, b, acc);
  }

  const int n = n0 + r;                     // C/D column per lane
  float cnt = 0.f;
#pragma unroll
  for (int v = 0; v < 8; ++v) {
    const int bi  = m0 + v + 8 * half;      // C/D row
    const int idx = bi * N_NEUR + n;
    float m_ = mem[idx];
    float s_ = ALPHA * syn[idx] + acc[v];
    const float reset = (m_ > 1.0f) ? 1.0f : 0.0f;
    m_ = BETA * m_ + s_ - reset;            // reset-by-subtraction, THRESHOLD=1
    const float sp = (m_ > 1.0f) ? 1.0f : 0.0f;
    syn[idx] = s_;
    mem[idx] = m_;
    post_tr[idx] = BETA_MINUS * post_tr[idx] + sp;
    spk[idx] = sp;
    cnt += sp;
  }
  atomicAdd(&counts[n], cnt);               // per-neuron spike totals
}

// ---------------------------------------------------------------------------
// stdp: dW = (A+ · spkᵀ·pre_tr − A− · post_trᵀ·x)/B ; W = clip(W+dW, 0, 1)
// 25x49 = 1225 16x16 tiles of W, one per wave, K = BATCH = 128
// ---------------------------------------------------------------------------
__global__ __launch_bounds__(256) void stdp_kernel(
    const float* __restrict__ x, float* __restrict__ W,
    const float* __restrict__ spk, const float* __restrict__ post_tr,
    const float* __restrict__ pre_tr) {
  const int wave = threadIdx.x >> 5;
  const int lane = threadIdx.x & 31;
  const int tile = blockIdx.x * 4 + wave;
  if (tile >= 25 * 49) return;              // wave-uniform guard
  const int n0 = (tile % 25) << 4;          // neuron rows of W
  const int i0 = (tile / 25) << 4;          // input cols of W
  const int r    = lane & 15;
  const int half = lane >> 4;

  v8f accP = {0.f, 0.f, 0.f, 0.f, 0.f, 0.f, 0.f, 0.f};
  v8f accD = {0.f, 0.f, 0.f, 0.f, 0.f, 0.f, 0.f, 0.f};
#pragma unroll 8
  for (int k = 0; k < BATCH; k += 4) {
    const int kk = k + 2 * half;
    // A[n][b] = spk[b*N + n] (implicit transpose); B[b][i] = pre_tr[b*IN + i]
    v2f aP = { spk[kk * N_NEUR + n0 + r],     spk[(kk + 1) * N_NEUR + n0 + r] };
    v2f bP = { pre_tr[kk * IN_DIM + i0 + r],  pre_tr[(kk + 1) * IN_DIM + i0 + r] };
    accP = wmma_f32_k4(aP, bP, accP);
    v2f aD = { post_tr[kk * N_NEUR + n0 + r], post_tr[(kk + 1) * N_NEUR + n0 + r] };
    v2f bD = { x[kk * IN_DIM + i0 + r],       x[(kk + 1) * IN_DIM + i0 + r] };
    accD = wmma_f32_k4(aD, bD, accD);
  }
#pragma unroll
  for (int v = 0; v < 8; ++v) {
    const int nn  = n0 + v + 8 * half;
    const int idx = nn * IN_DIM + i0 + r;
    float w = W[idx] + (A_PLUS * accP[v] - A_MINUS * accD[v]) * (1.0f / (float)BATCH);
    W[idx] = fminf(fmaxf(w, 0.0f), 1.0f);
  }
}

// ---------------------------------------------------------------------------
extern "C" void kernel_launch(void* const* d_in, const int* in_sizes, int n_in,
                              void* d_out, int out_size, void* d_ws, size_t ws_size,
                              hipStream_t stream) {
  const float* image = (const float*)d_in[0];   // [350,128,784]
  const float* W0    = (const float*)d_in[1];   // [400,784]

  float* Wout   = (float*)d_out;                       // 313600
  float* mem    = Wout + N_NEUR * IN_DIM;              //  51200
  float* syn    = mem + BATCH * N_NEUR;                //  51200
  float* counts = syn + BATCH * N_NEUR;                //    400
  float* pre    = (float*)d_ws;                        // 100352
  float* post   = pre + BATCH * IN_DIM;                //  51200
  float* spk    = post + BATCH * N_NEUR;               //  51200

  init_kernel<<<1627, 256, 0, stream>>>(W0, Wout, pre);

  for (int t = 0; t < T_STEPS; ++t) {
    const float* x_t = image + (size_t)t * BATCH * IN_DIM;
    fwd_kernel<<<442, 256, 0, stream>>>(x_t, Wout, syn, mem, post, spk, pre, counts);
    stdp_kernel<<<307, 256, 0, stream>>>(x_t, Wout, spk, post, pre);
  }

  (void)in_sizes; (void)n_in; (void)out_size; (void)ws_size;
}